// NHGCF_65910568124541
// MI455X (gfx1250) — compile-verified
//
#include <hip/hip_runtime.h>
#include <cstddef>
#include <cstdint>

// ---------------- problem constants (from reference) ----------------
#define NUU 100000
#define NII 50000
#define NEU 20000
#define NEI 20000
#define DD  64
#define HH  32
#define NNZ_U2I 2000000
#define NNZ_U2E 1000000
#define NNZ_I2E 1000000
#define NQ  4096
#define SLOPE 0.01f

static constexpr int N_U2E = NUU + NEU;                     // 120000
static constexpr int N_I2E = NII + NEI;                     // 70000
static constexpr int N_U2I = NUU + NII;                     // 150000
static constexpr int PAD_U2E = ((N_U2E + 127) / 128) * 128; // 120064
static constexpr int PAD_I2E = ((N_I2E + 127) / 128) * 128; // 70016
static constexpr int PAD_U2I = ((N_U2I + 127) / 128) * 128; // 150016

typedef float v2f __attribute__((ext_vector_type(2)));
typedef float v8f __attribute__((ext_vector_type(8)));

// LDS layout (floats). Weight region is overlaid by the output staging
// region after the K-loop (guarded by __syncthreads()).
static constexpr int LW   = 66;                 // padded stride of transposed W
static constexpr int OWT  = 0;                  // sWt  : 64*66 = 4224
static constexpr int OWI  = 64 * LW;            // sWi  : 4224
static constexpr int OBT  = 2 * 64 * LW;        // sbt  : 64
static constexpr int OBI  = OBT + 64;           // sbi  : 64
static constexpr int LSO  = 68;                 // padded stride of staging tile
static constexpr int LDSF = 8 * 16 * LSO;       // 8704 floats (> 8576 weight use)

// =====================================================================
// Dense GEMM via WMMA f32 16x16x4:
//   T = X @ Wt + bt ;  Q = (X*X) @ Wi + bi        (64x64 weights)
// Block = 256 threads = 8 wave32; each wave owns a 16-row strip and all
// 64 output columns (4 accumulator tiles per matrix).
//  - weights staged TRANSPOSED in LDS -> B fragment = one aligned b64 load
//  - bias pre-loaded into accumulators (no tail adds)
//  - output repacked via per-wave LDS tile -> coalesced b128 stores
// n_pad must be a multiple of 128; EXEC stays all-ones (WMMA requirement).
// =====================================================================
__global__ __launch_bounds__(256) void gemm_tq_kernel(
    const float* __restrict__ X,
    const float* __restrict__ Wt, const float* __restrict__ bt,
    const float* __restrict__ Wi, const float* __restrict__ bi,
    float* __restrict__ T, float* __restrict__ Q, int n_pad)
{
    __shared__ float lds[LDSF];
    (void)n_pad;

    const int tid = threadIdx.x;

    // ---- transpose-fill weights: sW[c*66 + k] = W[k][c] ----
    for (int i = tid * 4; i < 64 * 64; i += 256 * 4) {
        const int k = i >> 6;   // row of W (K index)
        const int c = i & 63;   // first of 4 columns
        const float4 wt = *(const float4*)&Wt[i];
        const float4 wi = *(const float4*)&Wi[i];
        lds[OWT + (c + 0) * LW + k] = wt.x;
        lds[OWT + (c + 1) * LW + k] = wt.y;
        lds[OWT + (c + 2) * LW + k] = wt.z;
        lds[OWT + (c + 3) * LW + k] = wt.w;
        lds[OWI + (c + 0) * LW + k] = wi.x;
        lds[OWI + (c + 1) * LW + k] = wi.y;
        lds[OWI + (c + 2) * LW + k] = wi.z;
        lds[OWI + (c + 3) * LW + k] = wi.w;
    }
    if (tid < 64) { lds[OBT + tid] = bt[tid]; lds[OBI + tid] = bi[tid]; }
    __syncthreads();

    const int wave = tid >> 5;
    const int lane = tid & 31;
    const int rowBase = blockIdx.x * 128 + wave * 16;

    // A-fragment addressing: lanes 0-15 hold K={0,1}, lanes 16-31 K={2,3}
    const int aRow = rowBase + (lane & 15);
    const int aK   = (lane >> 4) * 2;
    const float* xrow = X + (size_t)aRow * DD + aK;

    const int bn  = (lane & 15);       // output column within 16-wide tile
    const int bkh = (lane >> 4) * 2;   // K sub-offset for B fragment

    // ---- accumulators pre-loaded with bias (bias depends on col only) ----
    v8f accT[4], accQ[4];
#pragma unroll
    for (int j = 0; j < 4; ++j) {
        const float bTa = lds[OBT + 16 * j + bn];
        const float bQa = lds[OBI + 16 * j + bn];
#pragma unroll
        for (int r = 0; r < 8; ++r) { accT[j][r] = bTa; accQ[j][r] = bQa; }
    }

    for (int kk = 0; kk < 16; ++kk) {
        const v2f a  = *(const v2f*)(xrow + 4 * kk);
        const v2f aq = a * a;          // elementwise square for (X*X)@Wi
        const int bk = 4 * kk + bkh;
#pragma unroll
        for (int j = 0; j < 4; ++j) {
            const int c = 16 * j + bn;
            const v2f bT = *(const v2f*)&lds[OWT + c * LW + bk];
            const v2f bQ = *(const v2f*)&lds[OWI + c * LW + bk];
            accT[j] = __builtin_amdgcn_wmma_f32_16x16x4_f32(
                false, a,  false, bT, (short)0, accT[j], false, false);
            accQ[j] = __builtin_amdgcn_wmma_f32_16x16x4_f32(
                false, aq, false, bQ, (short)0, accQ[j], false, false);
        }
    }

    // ---- repack 16x64 tile through LDS, then coalesced b128 stores ----
    __syncthreads();                       // weights dead; overlay staging
    float* so = lds + wave * (16 * LSO);   // per-wave region, in-order DS
    const int  rOff = (lane >> 4) * 8;
    const size_t gBase = (size_t)rowBase * DD;

    // phase T
#pragma unroll
    for (int j = 0; j < 4; ++j)
#pragma unroll
        for (int r = 0; r < 8; ++r)
            so[(rOff + r) * LSO + 16 * j + bn] = accT[j][r];
    __asm volatile("" ::: "memory");
#pragma unroll
    for (int it = 0; it < 8; ++it) {
        const int flat = it * 128 + lane * 4;
        const int row = flat >> 6, col = flat & 63;
        *(float4*)&T[gBase + flat] = *(const float4*)&so[row * LSO + col];
    }
    __asm volatile("" ::: "memory");

    // phase Q (DS pipeline is in-order per wave: safe to overwrite)
#pragma unroll
    for (int j = 0; j < 4; ++j)
#pragma unroll
        for (int r = 0; r < 8; ++r)
            so[(rOff + r) * LSO + 16 * j + bn] = accQ[j][r];
    __asm volatile("" ::: "memory");
#pragma unroll
    for (int it = 0; it < 8; ++it) {
        const int flat = it * 128 + lane * 4;
        const int row = flat >> 6, col = flat & 63;
        *(float4*)&Q[gBase + flat] = *(const float4*)&so[row * LSO + col];
    }
}

// =====================================================================
// SpMM scatter: dst[row] += val * src[col]  (hardware f32 atomics).
// Two threads per edge (32 contiguous columns each) -> coalesced.
// =====================================================================
__global__ __launch_bounds__(256) void spmm_atomic_kernel(
    const int* __restrict__ rows, const int* __restrict__ cols,
    const float* __restrict__ vals, int nnz,
    const float* __restrict__ src, float* __restrict__ dst)
{
    const int idx  = blockIdx.x * blockDim.x + threadIdx.x;
    const int e    = idx >> 1;
    if (e >= nnz) return;
    const int half = (idx & 1) * 32;
    const float v  = vals[e];
    const float* s = src + (size_t)cols[e] * DD + half;
    float*       d = dst + (size_t)rows[e] * DD + half;
#pragma unroll
    for (int k = 0; k < 32; k += 4) {
        const float4 sv = *(const float4*)(s + k);
        unsafeAtomicAdd(d + k + 0, v * sv.x);
        unsafeAtomicAdd(d + k + 1, v * sv.y);
        unsafeAtomicAdd(d + k + 2, v * sv.z);
        unsafeAtomicAdd(d + k + 3, v * sv.w);
    }
}

// =====================================================================
// 2-way relation attention + leaky ReLU (one thread per node; tiny work).
//   w_j = tanh(z_j @ W1 + b1) @ w2 ; beta = softmax(w) ;
//   out = leaky(beta0*z0 + beta1*z1)
// =====================================================================
__global__ __launch_bounds__(256) void attn_leaky_kernel(
    const float* __restrict__ z0s, const float* __restrict__ z1s,
    const float* __restrict__ W1, const float* __restrict__ b1,
    const float* __restrict__ w2, float* __restrict__ out, int n)
{
    const int i = blockIdx.x * blockDim.x + threadIdx.x;
    if (i >= n) return;
    const float* z0 = z0s + (size_t)i * DD;
    const float* z1 = z1s + (size_t)i * DD;

    float w[2];
#pragma unroll
    for (int j = 0; j < 2; ++j) {
        const float* z = j ? z1 : z0;
        float s[HH];
#pragma unroll
        for (int h = 0; h < HH; ++h) s[h] = b1[h];
        for (int k = 0; k < DD; ++k) {
            const float zk = z[k];
#pragma unroll
            for (int h = 0; h < HH; ++h) s[h] += zk * W1[k * HH + h];
        }
        float acc = 0.f;
#pragma unroll
        for (int h = 0; h < HH; ++h) acc += tanhf(s[h]) * w2[h];
        w[j] = acc;
    }
    const float m  = fmaxf(w[0], w[1]);
    const float e0 = __expf(w[0] - m);
    const float e1 = __expf(w[1] - m);
    const float inv = 1.f / (e0 + e1);
    const float b0 = e0 * inv, bB = e1 * inv;
#pragma unroll
    for (int k = 0; k < DD; k += 4) {
        const float4 a = *(const float4*)(z0 + k);
        const float4 b = *(const float4*)(z1 + k);
        float4 o;
        o.x = b0 * a.x + bB * b.x; o.y = b0 * a.y + bB * b.y;
        o.z = b0 * a.z + bB * b.z; o.w = b0 * a.w + bB * b.w;
        o.x = o.x >= 0.f ? o.x : SLOPE * o.x;
        o.y = o.y >= 0.f ? o.y : SLOPE * o.y;
        o.z = o.z >= 0.f ? o.z : SLOPE * o.z;
        o.w = o.w >= 0.f ? o.w : SLOPE * o.w;
        *(float4*)(out + (size_t)i * DD + k) = o;
    }
}

__global__ __launch_bounds__(256) void leaky_kernel(
    const float* __restrict__ in, float* __restrict__ out, size_t n)
{
    const size_t i = (size_t)blockIdx.x * blockDim.x + threadIdx.x;
    if (i >= n) return;
    const float v = in[i];
    out[i] = v >= 0.f ? v : SLOPE * v;
}

// =====================================================================
// Final: out[p] = dot(concat(uE0,uE1,uE2)[u], concat(iE0,iE1,iE2)[it])
// =====================================================================
__global__ __launch_bounds__(256) void dot_kernel(
    const float* __restrict__ uE0, const float* __restrict__ uE1,
    const float* __restrict__ uE2,
    const float* __restrict__ iE0, const float* __restrict__ iE1,
    const float* __restrict__ iE2,
    const int* __restrict__ uIdx, const int* __restrict__ iIdx,
    float* __restrict__ out, int n)
{
    const int p = blockIdx.x * blockDim.x + threadIdx.x;
    if (p >= n) return;
    const size_t u  = (size_t)uIdx[p] * DD;
    const size_t it = (size_t)iIdx[p] * DD;
    float acc = 0.f;
#pragma unroll 8
    for (int k = 0; k < DD; ++k) {
        acc += uE0[u + k] * iE0[it + k];
        acc += uE1[u + k] * iE1[it + k];
        acc += uE2[u + k] * iE2[it + k];
    }
    out[p] = acc;
}

// =====================================================================
// Host orchestration
// =====================================================================
extern "C" void kernel_launch(void* const* d_in, const int* in_sizes, int n_in,
                              void* d_out, int out_size, void* d_ws, size_t ws_size,
                              hipStream_t stream)
{
    (void)in_sizes; (void)n_in; (void)out_size; (void)ws_size;

    // ---- input map (setup_inputs dict order, recursively flattened) ----
    const float* uEmbd  = (const float*)d_in[0];
    const float* iEmbd  = (const float*)d_in[1];
    const float* ueEmbd = (const float*)d_in[2];
    const float* ieEmbd = (const float*)d_in[3];
    // per layer l, base = 4 + 18*l:
    //  +0..3  u2i {Wt,bt,Wi,bi}; +4..7 u2e; +8..11 i2e;
    //  +12..14 u_attn {W1,b1,w2}; +15..17 i_attn {W1,b1,w2}
    const int* g_rows[3]; const int* g_cols[3]; const float* g_vals[3];
    g_rows[0] = (const int*)d_in[40]; g_cols[0] = (const int*)d_in[41]; g_vals[0] = (const float*)d_in[42]; // u2i
    g_rows[1] = (const int*)d_in[44]; g_cols[1] = (const int*)d_in[45]; g_vals[1] = (const float*)d_in[46]; // u2e
    g_rows[2] = (const int*)d_in[48]; g_cols[2] = (const int*)d_in[49]; g_vals[2] = (const float*)d_in[50]; // i2e
    const int* userIdx = (const int*)d_in[52];
    const int* itemIdx = (const int*)d_in[53];
    float* out = (float*)d_out;

    // ---- workspace layout (floats) ----
    float* ws = (float*)d_ws;
    size_t o = 0;
    float* X   = ws + o; o += (size_t)PAD_U2I * DD;
    float* T   = ws + o; o += (size_t)PAD_U2I * DD;
    float* Q   = ws + o; o += (size_t)PAD_U2I * DD;
    float* R1  = ws + o; o += (size_t)PAD_U2E * DD;  // gcn(u2e) result
    float* R2  = ws + o; o += (size_t)PAD_I2E * DD;  // gcn(i2e) result
    float* R3  = ws + o; o += (size_t)PAD_U2I * DD;  // gcn(u2i) result
    float* ueB = ws + o; o += (size_t)NEU * DD;
    float* ieB = ws + o; o += (size_t)NEI * DD;
    float* uL[2]; uL[0] = ws + o; o += (size_t)NUU * DD; uL[1] = ws + o; o += (size_t)NUU * DD;
    float* iL[2]; iL[0] = ws + o; o += (size_t)NII * DD; iL[1] = ws + o; o += (size_t)NII * DD;

    const size_t rowB = (size_t)DD * sizeof(float);
    const dim3 B256(256);

    const float* u_f  = uEmbd;
    const float* i_f  = iEmbd;
    const float* ue_f = ueEmbd;
    const float* ie_f = ieEmbd;

    for (int l = 0; l < 2; ++l) {
        const int base = 4 + 18 * l;
        const float* Wt_u2i = (const float*)d_in[base + 0];
        const float* bt_u2i = (const float*)d_in[base + 1];
        const float* Wi_u2i = (const float*)d_in[base + 2];
        const float* bi_u2i = (const float*)d_in[base + 3];
        const float* Wt_u2e = (const float*)d_in[base + 4];
        const float* bt_u2e = (const float*)d_in[base + 5];
        const float* Wi_u2e = (const float*)d_in[base + 6];
        const float* bi_u2e = (const float*)d_in[base + 7];
        const float* Wt_i2e = (const float*)d_in[base + 8];
        const float* bt_i2e = (const float*)d_in[base + 9];
        const float* Wi_i2e = (const float*)d_in[base + 10];
        const float* bi_i2e = (const float*)d_in[base + 11];
        const float* uA_W1  = (const float*)d_in[base + 12];
        const float* uA_b1  = (const float*)d_in[base + 13];
        const float* uA_w2  = (const float*)d_in[base + 14];
        const float* iA_W1  = (const float*)d_in[base + 15];
        const float* iA_b1  = (const float*)d_in[base + 16];
        const float* iA_w2  = (const float*)d_in[base + 17];

        // ---- gcn_cell(u2e) on concat(u_f, ue_f) -> R1 ----
        hipMemcpyAsync(X, u_f, (size_t)NUU * rowB, hipMemcpyDeviceToDevice, stream);
        hipMemcpyAsync(X + (size_t)NUU * DD, ue_f, (size_t)NEU * rowB, hipMemcpyDeviceToDevice, stream);
        gemm_tq_kernel<<<PAD_U2E / 128, B256, 0, stream>>>(X, Wt_u2e, bt_u2e, Wi_u2e, bi_u2e, T, Q, PAD_U2E);
        hipMemcpyAsync(R1, T, (size_t)N_U2E * rowB, hipMemcpyDeviceToDevice, stream); // "+ t" term
        spmm_atomic_kernel<<<(2 * NNZ_U2E + 255) / 256, B256, 0, stream>>>(g_rows[1], g_cols[1], g_vals[1], NNZ_U2E, T, R1);
        spmm_atomic_kernel<<<(2 * NNZ_U2E + 255) / 256, B256, 0, stream>>>(g_rows[1], g_cols[1], g_vals[1], NNZ_U2E, Q, R1);

        // ---- gcn_cell(i2e) on concat(i_f, ie_f) -> R2 ----
        hipMemcpyAsync(X, i_f, (size_t)NII * rowB, hipMemcpyDeviceToDevice, stream);
        hipMemcpyAsync(X + (size_t)NII * DD, ie_f, (size_t)NEI * rowB, hipMemcpyDeviceToDevice, stream);
        gemm_tq_kernel<<<PAD_I2E / 128, B256, 0, stream>>>(X, Wt_i2e, bt_i2e, Wi_i2e, bi_i2e, T, Q, PAD_I2E);
        hipMemcpyAsync(R2, T, (size_t)N_I2E * rowB, hipMemcpyDeviceToDevice, stream);
        spmm_atomic_kernel<<<(2 * NNZ_I2E + 255) / 256, B256, 0, stream>>>(g_rows[2], g_cols[2], g_vals[2], NNZ_I2E, T, R2);
        spmm_atomic_kernel<<<(2 * NNZ_I2E + 255) / 256, B256, 0, stream>>>(g_rows[2], g_cols[2], g_vals[2], NNZ_I2E, Q, R2);

        // ---- gcn_cell(u2i) on concat(u_f, i_f) -> R3 ----
        hipMemcpyAsync(X, u_f, (size_t)NUU * rowB, hipMemcpyDeviceToDevice, stream);
        hipMemcpyAsync(X + (size_t)NUU * DD, i_f, (size_t)NII * rowB, hipMemcpyDeviceToDevice, stream);
        gemm_tq_kernel<<<PAD_U2I / 128, B256, 0, stream>>>(X, Wt_u2i, bt_u2i, Wi_u2i, bi_u2i, T, Q, PAD_U2I);
        hipMemcpyAsync(R3, T, (size_t)N_U2I * rowB, hipMemcpyDeviceToDevice, stream);
        spmm_atomic_kernel<<<(2 * NNZ_U2I + 255) / 256, B256, 0, stream>>>(g_rows[0], g_cols[0], g_vals[0], NNZ_U2I, T, R3);
        spmm_atomic_kernel<<<(2 * NNZ_U2I + 255) / 256, B256, 0, stream>>>(g_rows[0], g_cols[0], g_vals[0], NNZ_U2I, Q, R3);

        // ---- attention + leaky ----
        attn_leaky_kernel<<<(NUU + 255) / 256, B256, 0, stream>>>(R1, R3, uA_W1, uA_b1, uA_w2, uL[l], NUU);
        attn_leaky_kernel<<<(NII + 255) / 256, B256, 0, stream>>>(R2, R3 + (size_t)NUU * DD, iA_W1, iA_b1, iA_w2, iL[l], NII);
        leaky_kernel<<<((size_t)NEU * DD + 255) / 256, B256, 0, stream>>>(R1 + (size_t)NUU * DD, ueB, (size_t)NEU * DD);
        leaky_kernel<<<((size_t)NEI * DD + 255) / 256, B256, 0, stream>>>(R2 + (size_t)NII * DD, ieB, (size_t)NEI * DD);

        u_f = uL[l]; i_f = iL[l]; ue_f = ueB; ie_f = ieB;
    }

    // ---- final gathered dot products ----
    dot_kernel<<<(NQ + 255) / 256, B256, 0, stream>>>(
        uEmbd, uL[0], uL[1], iEmbd, iL[0], iL[1], userIdx, itemIdx, out, NQ);
}